// QuPoolLast_712964571926
// MI455X (gfx1250) — compile-verified
//
#include <hip/hip_runtime.h>
#include <hip/hip_bf16.h>

typedef __attribute__((ext_vector_type(2))) float v2f;
typedef __attribute__((ext_vector_type(8))) float v8f;

// ---------------------------------------------------------------------------
// Setup: build U from weight (10-gate circuit), fold U (.) U^dagger into a
// real 16x32 operator W whose 32 columns match the interleaved complex64
// output layout.  W[jk][2*(i*4+l)+p] = {Re,Im}( U[i][j] * conj(U[l][k]) ).
// ---------------------------------------------------------------------------
struct cpx { float re, im; };
__device__ __forceinline__ cpx cmul(cpx a, cpx b) {
  return { a.re * b.re - a.im * b.im, a.re * b.im + a.im * b.re };
}
__device__ __forceinline__ cpx cadd(cpx a, cpx b) { return { a.re + b.re, a.im + b.im }; }
__device__ __forceinline__ cpx cconj(cpx a) { return { a.re, -a.im }; }

__device__ void mat4_apply(cpx U[4][4], const cpx G[4][4]) {  // U = G @ U
  cpx T[4][4];
  for (int i = 0; i < 4; ++i)
    for (int j = 0; j < 4; ++j) {
      cpx acc = { 0.f, 0.f };
      for (int k = 0; k < 4; ++k) acc = cadd(acc, cmul(G[i][k], U[k][j]));
      T[i][j] = acc;
    }
  for (int i = 0; i < 4; ++i)
    for (int j = 0; j < 4; ++j) U[i][j] = T[i][j];
}

__device__ void kron_q0(cpx M[4][4], const cpx g[2][2]) {  // kron(g, I2)
  for (int i = 0; i < 2; ++i)
    for (int j = 0; j < 2; ++j)
      for (int a = 0; a < 2; ++a)
        for (int b = 0; b < 2; ++b)
          M[2 * i + a][2 * j + b] = (a == b) ? g[i][j] : cpx{ 0.f, 0.f };
}
__device__ void kron_q1(cpx M[4][4], const cpx g[2][2]) {  // kron(I2, g)
  for (int a = 0; a < 2; ++a)
    for (int b = 0; b < 2; ++b)
      for (int i = 0; i < 2; ++i)
        for (int j = 0; j < 2; ++j)
          M[2 * a + i][2 * b + j] = (a == b) ? g[i][j] : cpx{ 0.f, 0.f };
}
__device__ void g_rx(cpx g[2][2], float t) {
  float c = cosf(0.5f * t), s = sinf(0.5f * t);
  g[0][0] = { c, 0.f }; g[0][1] = { 0.f, -s };
  g[1][0] = { 0.f, -s }; g[1][1] = { c, 0.f };
}
__device__ void g_ry(cpx g[2][2], float t) {
  float c = cosf(0.5f * t), s = sinf(0.5f * t);
  g[0][0] = { c, 0.f }; g[0][1] = { -s, 0.f };
  g[1][0] = { s, 0.f }; g[1][1] = { c, 0.f };
}
__device__ void g_rz(cpx g[2][2], float t) {
  float c = cosf(0.5f * t), s = sinf(0.5f * t);
  g[0][0] = { c, -s };  g[0][1] = { 0.f, 0.f };
  g[1][0] = { 0.f, 0.f }; g[1][1] = { c, s };
}

__global__ void qu_setup_kernel(const float* __restrict__ weight,
                                float* __restrict__ W) {
  if (threadIdx.x != 0 || blockIdx.x != 0) return;
  const float WMUL = 0.63245553203367587f;  // sqrt(2)*5^-0.5
  float w[6];
  for (int i = 0; i < 6; ++i) w[i] = weight[i] * WMUL;

  cpx U[4][4];
  for (int i = 0; i < 4; ++i)
    for (int j = 0; j < 4; ++j) U[i][j] = { (i == j) ? 1.f : 0.f, 0.f };

  cpx g[2][2], G[4][4];
  g_rx(g, w[0]); kron_q0(G, g); mat4_apply(U, G);
  g_ry(g, w[1]); kron_q0(G, g); mat4_apply(U, G);
  g_rz(g, w[2]); kron_q0(G, g); mat4_apply(U, G);
  g_rx(g, w[3]); kron_q1(G, g); mat4_apply(U, G);
  g_ry(g, w[4]); kron_q1(G, g); mat4_apply(U, G);
  g_rz(g, w[5]); kron_q1(G, g); mat4_apply(U, G);
  // CNOT (control q0, target q1)
  for (int i = 0; i < 4; ++i)
    for (int j = 0; j < 4; ++j) G[i][j] = { 0.f, 0.f };
  G[0][0] = { 1.f, 0.f }; G[1][1] = { 1.f, 0.f };
  G[2][3] = { 1.f, 0.f }; G[3][2] = { 1.f, 0.f };
  mat4_apply(U, G);
  g_rz(g, -w[5]); kron_q1(G, g); mat4_apply(U, G);
  g_ry(g, -w[4]); kron_q1(G, g); mat4_apply(U, G);
  g_rx(g, -w[3]); kron_q1(G, g); mat4_apply(U, G);

  // Fold into real 16x32 operator, row = (j*4+k), col = 2*(i*4+l)+{re,im}
  for (int j = 0; j < 4; ++j)
    for (int k = 0; k < 4; ++k)
      for (int i = 0; i < 4; ++i)
        for (int l = 0; l < 4; ++l) {
          cpx t = cmul(U[i][j], cconj(U[l][k]));
          W[(j * 4 + k) * 32 + (i * 4 + l) * 2 + 0] = t.re;
          W[(j * 4 + k) * 32 + (i * 4 + l) * 2 + 1] = t.im;
        }
}

// ---------------------------------------------------------------------------
// Main streaming kernel: out32[B x 32] = X16[B x 32... wait 16] @ W[16 x 32]
// One wave handles 16 batch elements per tile via 8 chained
// v_wmma_f32_16x16x4_f32 ops (2 output tiles x K=16 in chunks of 4).
// ---------------------------------------------------------------------------
__global__ void qu_wmma_kernel(const float* __restrict__ x,
                               const float* __restrict__ W,
                               float* __restrict__ out,
                               int ntiles) {
  const int lane = threadIdx.x & 31;
  const int wid  = blockIdx.x * (blockDim.x >> 5) + (threadIdx.x >> 5);
  const int nw   = gridDim.x * (blockDim.x >> 5);
  const int r  = lane & 15;   // row-within-tile / column index
  const int hi = lane >> 4;   // half-wave select (K pair / row+8)

  // B operands: 4 K-chunks x 2 column halves of W, resident in VGPRs.
  // B-matrix 4x16 layout: VGPR v, lanes 0-15 -> K = v, lanes 16-31 -> K = v+2.
  v2f wb[4][2];
#pragma unroll
  for (int c = 0; c < 4; ++c)
#pragma unroll
    for (int h = 0; h < 2; ++h) {
      int col = r + h * 16;
      wb[c][h][0] = W[(c * 4 + hi * 2 + 0) * 32 + col];
      wb[c][h][1] = W[(c * 4 + hi * 2 + 1) * 32 + col];
    }

  for (int t = wid; t < ntiles; t += nw) {
    const float* xp = x + (long)t * 256;  // 16 elements x 16 floats

    // A operands: 16x4 f32 layout -> lane m (0-15) holds K=0,1; lane m+16
    // holds K=2,3.  One coalesced b64 load per K-chunk.
    v2f a[4];
#pragma unroll
    for (int c = 0; c < 4; ++c)
      a[c] = *(const v2f*)(xp + r * 16 + c * 4 + hi * 2);

    v8f dl = {};  // output columns 0..15
    v8f dr = {};  // output columns 16..31
#pragma unroll
    for (int c = 0; c < 4; ++c) {
      dl = __builtin_amdgcn_wmma_f32_16x16x4_f32(
          false, a[c], false, wb[c][0], (short)0, dl, false, false);
      dr = __builtin_amdgcn_wmma_f32_16x16x4_f32(
          false, a[c], false, wb[c][1], (short)0, dr, false, false);
    }

    // D layout: VGPR v, lanes 0-15 -> row v, col=lane; lanes 16-31 -> row v+8.
    float* op = out + (long)t * 512;
#pragma unroll
    for (int v = 0; v < 8; ++v) {
      int row = v + hi * 8;
      op[row * 32 + r]      = dl[v];
      op[row * 32 + 16 + r] = dr[v];
    }
  }
}

// Scalar tail (B % 16 != 0) — unused for B = 1M but kept for safety.
__global__ void qu_tail_kernel(const float* __restrict__ x,
                               const float* __restrict__ W,
                               float* __restrict__ out,
                               int start, int n) {
  int b = start + blockIdx.x * blockDim.x + threadIdx.x;
  if (b >= n) return;
  const float* xp = x + (long)b * 16;
  float xv[16];
  for (int i = 0; i < 16; ++i) xv[i] = xp[i];
  float* op = out + (long)b * 32;
  for (int c = 0; c < 32; ++c) {
    float acc = 0.f;
    for (int jk = 0; jk < 16; ++jk) acc += xv[jk] * W[jk * 32 + c];
    op[c] = acc;
  }
}

extern "C" void kernel_launch(void* const* d_in, const int* in_sizes, int n_in,
                              void* d_out, int out_size, void* d_ws, size_t ws_size,
                              hipStream_t stream) {
  const float* x      = (const float*)d_in[0];
  const float* weight = (const float*)d_in[1];
  float* out = (float*)d_out;
  float* W   = (float*)d_ws;  // 16 x 32 fp32 operator (2 KB)

  long n = (long)in_sizes[0] / 16;          // batch count
  qu_setup_kernel<<<1, 32, 0, stream>>>(weight, W);

  long ntiles = n / 16;
  if (ntiles > 0) {
    const int waves_per_block = 8;          // 256 threads
    long blocks = (ntiles + waves_per_block - 1) / waves_per_block;
    if (blocks > 32768) blocks = 32768;
    qu_wmma_kernel<<<(int)blocks, 256, 0, stream>>>(x, W, out, (int)ntiles);
  }
  long rem = n - ntiles * 16;
  if (rem > 0) {
    qu_tail_kernel<<<(int)((rem + 255) / 256), 256, 0, stream>>>(
        x, W, out, (int)(ntiles * 16), (int)n);
  }
}